// MultimodalFusion_3908420240286
// MI455X (gfx1250) — compile-verified
//
#include <hip/hip_runtime.h>

#define NN 50000
#define NE 800000
#define DIM 64
#define NH 4
#define DH 16

typedef float v2f __attribute__((ext_vector_type(2)));
typedef float v8f __attribute__((ext_vector_type(8)));

// ---------------------------------------------------------------------------
// out[nrows,64] = A[nrows,64] @ W[64,64]  via V_WMMA_F32_16X16X4_F32
// block = 128 threads (4 waves); wave w handles column tile w (16 cols).
// grid.x = nrows/16 row tiles. nrows must be a multiple of 16 (50000 is).
// ---------------------------------------------------------------------------
__global__ void gat_gemm64_wmma(const float* __restrict__ A,
                                const float* __restrict__ W,
                                float* __restrict__ out) {
  const int wave  = threadIdx.x >> 5;     // 0..3 : column tile
  const int lane  = threadIdx.x & 31;
  const int row0  = blockIdx.x * 16;
  const int m     = lane & 15;            // row within tile (A) / col within tile (B,D)
  const int khalf = lane >> 4;            // 0: K={0,1}, 1: K={2,3}
  const int n0    = wave * 16;

  v8f acc = {};
  for (int kk = 0; kk < DIM; kk += 4) {
    const int ka = kk + 2 * khalf;
    v2f a, b;
    // A 16x4 tile: lane holds row m, K = ka, ka+1 (adjacent -> 64-bit load)
    const float* ap = A + (size_t)(row0 + m) * DIM + ka;
    a.x = ap[0];
    a.y = ap[1];
    // B 4x16 tile: lane holds col n0+m, K = ka, ka+1
    b.x = W[(size_t)ka * DIM + n0 + m];
    b.y = W[(size_t)(ka + 1) * DIM + n0 + m];
    acc = __builtin_amdgcn_wmma_f32_16x16x4_f32(
        /*neg_a=*/false, a, /*neg_b=*/false, b,
        /*c_mod=*/(short)0, acc, /*reuse_a=*/false, /*reuse_b=*/false);
  }
  // D 16x16: lane holds col n0+m; VGPR r -> row r + 8*khalf
#pragma unroll
  for (int r = 0; r < 8; ++r) {
    const int mrow = r + 8 * khalf;
    out[(size_t)(row0 + mrow) * DIM + n0 + m] = acc[r];
  }
}

// ---------------------------------------------------------------------------
// Workspace init (harness poisons once; we must re-init every call).
// ---------------------------------------------------------------------------
__global__ void gat_init(float* __restrict__ mbuf, float* __restrict__ denom,
                         float* __restrict__ agg) {
  const int i = blockIdx.x * blockDim.x + threadIdx.x;
  if (i < NN * NH) {
    mbuf[i]  = -1e9f;   // mirrors reference's max(seg_max, -1e9)
    denom[i] = 0.0f;
  }
  if (i < NN * DIM) agg[i] = 0.0f;
}

// Order-consistent float atomic max via int/uint monotone mapping.
__device__ __forceinline__ void atomicMaxF(float* addr, float val) {
  if (val >= 0.0f)
    atomicMax((int*)addr, __float_as_int(val));
  else
    atomicMin((unsigned int*)addr, __float_as_uint(val));
}

// ---------------------------------------------------------------------------
// Per-(edge,head) logits + scatter max onto destination nodes.
// ---------------------------------------------------------------------------
__global__ void gat_edge_scores(const float* __restrict__ q,
                                const float* __restrict__ k,
                                const int* __restrict__ src,
                                const int* __restrict__ dst,
                                float* __restrict__ scores,
                                float* __restrict__ mbuf) {
  const int i = blockIdx.x * blockDim.x + threadIdx.x;
  if (i >= NE * NH) return;
  const int e = i >> 2;
  const int h = i & 3;
  const int s = src[e];
  const int d = dst[e];
  const float4* qp = (const float4*)(q + (size_t)d * DIM + h * DH);
  const float4* kp = (const float4*)(k + (size_t)s * DIM + h * DH);
  float acc = 0.0f;
#pragma unroll
  for (int j = 0; j < 4; ++j) {
    const float4 a = qp[j];
    const float4 b = kp[j];
    acc += a.x * b.x + a.y * b.y + a.z * b.z + a.w * b.w;
  }
  acc *= 0.25f;  // 1/sqrt(Dh), Dh = 16
  scores[i] = acc;
  atomicMaxF(&mbuf[d * NH + h], acc);
}

// ---------------------------------------------------------------------------
// ex = exp(score - m[dst]); scatter-sum denominators.
// ---------------------------------------------------------------------------
__global__ void gat_edge_exp(const int* __restrict__ dst,
                             const float* __restrict__ mbuf,
                             float* __restrict__ scores,
                             float* __restrict__ denom) {
  const int i = blockIdx.x * blockDim.x + threadIdx.x;
  if (i >= NE * NH) return;
  const int e = i >> 2;
  const int h = i & 3;
  const int d = dst[e];
  const float ex = expf(scores[i] - mbuf[d * NH + h]);
  scores[i] = ex;  // overwrite logits with numerators
  atomicAdd(&denom[d * NH + h], ex);
}

// ---------------------------------------------------------------------------
// agg[dst, :] += alpha * v[src, :]; one thread per (edge, dim) so adjacent
// lanes issue atomics to adjacent addresses (coalesced atomic batches).
// ---------------------------------------------------------------------------
__global__ void gat_edge_agg(const int* __restrict__ src,
                             const int* __restrict__ dst,
                             const float* __restrict__ v,
                             const float* __restrict__ scores,
                             const float* __restrict__ denom,
                             float* __restrict__ agg) {
  const int i = blockIdx.x * blockDim.x + threadIdx.x;
  if (i >= NE * DIM) return;
  const int e    = i >> 6;
  const int dcol = i & 63;
  const int h    = dcol >> 4;
  const int s = src[e];
  const int d = dst[e];
  const float alpha = scores[e * NH + h] / (denom[d * NH + h] + 1e-9f);
  atomicAdd(&agg[(size_t)d * DIM + dcol], alpha * v[(size_t)s * DIM + dcol]);
}

// ---------------------------------------------------------------------------
extern "C" void kernel_launch(void* const* d_in, const int* in_sizes, int n_in,
                              void* d_out, int out_size, void* d_ws, size_t ws_size,
                              hipStream_t stream) {
  const float* x  = (const float*)d_in[0];
  const float* Wq = (const float*)d_in[1];
  const float* Wk = (const float*)d_in[2];
  const float* Wv = (const float*)d_in[3];
  const float* Wo = (const float*)d_in[4];
  const int* src  = (const int*)d_in[5];
  const int* dst  = (const int*)d_in[6];
  float* out = (float*)d_out;

  // Workspace carve-up (floats): q,k,v [N,64]; scores [E,4]; m,denom [N,4]; agg [N,64]
  float* q      = (float*)d_ws;
  float* kbuf   = q + (size_t)NN * DIM;
  float* vbuf   = kbuf + (size_t)NN * DIM;
  float* scores = vbuf + (size_t)NN * DIM;
  float* mbuf   = scores + (size_t)NE * NH;
  float* denom  = mbuf + (size_t)NN * NH;
  float* agg    = denom + (size_t)NN * NH;

  const int rowTiles = NN / 16;  // 3125

  // 1. init m/denom/agg
  {
    const int n = NN * DIM;
    gat_init<<<(n + 255) / 256, 256, 0, stream>>>(mbuf, denom, agg);
  }
  // 2-4. q,k,v projections (WMMA f32)
  gat_gemm64_wmma<<<rowTiles, 128, 0, stream>>>(x, Wq, q);
  gat_gemm64_wmma<<<rowTiles, 128, 0, stream>>>(x, Wk, kbuf);
  gat_gemm64_wmma<<<rowTiles, 128, 0, stream>>>(x, Wv, vbuf);
  // 5. edge logits + scatter max
  {
    const int n = NE * NH;
    gat_edge_scores<<<(n + 255) / 256, 256, 0, stream>>>(q, kbuf, src, dst, scores, mbuf);
  }
  // 6. exp + scatter denom
  {
    const int n = NE * NH;
    gat_edge_exp<<<(n + 255) / 256, 256, 0, stream>>>(dst, mbuf, scores, denom);
  }
  // 7. weighted aggregation
  {
    const int n = NE * DIM;
    gat_edge_agg<<<(n + 255) / 256, 256, 0, stream>>>(src, dst, vbuf, scores, denom, agg);
  }
  // 8. output projection -> d_out
  gat_gemm64_wmma<<<rowTiles, 128, 0, stream>>>(agg, Wo, out);
}